// _BatchNorm_27676769255584
// MI455X (gfx1250) — compile-verified
//
#include <hip/hip_runtime.h>

// Complex BatchNorm (training stats), [B,C,H,W] = [32,256,64,64] fp32.
// Two-pass streaming design, bandwidth-bound (~768 MiB total traffic).

typedef float v2f __attribute__((ext_vector_type(2)));
typedef float v4f __attribute__((ext_vector_type(4)));
typedef float v8f __attribute__((ext_vector_type(8)));

#define C_CH    256
#define BATCH   32
#define HW      4096
#define F4ROW   (HW / 4)          // 1024 float4 per (b,c) row
#define SPLIT   16                // stats blocks per channel
#define RPB     (BATCH / SPLIT)   // 2 rows per stats block
#define THREADS 256
#define NWAVES  (THREADS / 32)
#define EPSV    1e-5f

// ---------------------------------------------------------------------------
// Wave-level sum of one fp32 value per lane via V_WMMA_F32_16X16X4_F32.
// A (16x4 fp32, documented layout): lanes 0-15 hold {K0,K1}, lanes 16-31 hold
// {K2,K3} for row M = lane%16.  With A = {s, 0} and B = ones, D row m equals
// s[m] + s[m+16].  Rows 0..7 live in D v0..v7 (lanes 0-15), rows 8..15 in
// v0..v7 (lanes 16-31), so sum(v0..v7) at lane0 + lane16 = full 32-lane sum.
// ---------------------------------------------------------------------------
__device__ __forceinline__ float wave_sum(float s) {
#if defined(__gfx1250__) && __has_builtin(__builtin_amdgcn_wmma_f32_16x16x4_f32)
  v2f a; a.x = s;    a.y = 0.0f;
  v2f b; b.x = 1.0f; b.y = 1.0f;
  v8f c = {};
  v8f d = __builtin_amdgcn_wmma_f32_16x16x4_f32(
      /*neg_a=*/false, a, /*neg_b=*/false, b,
      /*c_mod=*/(short)0, c, /*reuse_a=*/false, /*reuse_b=*/false);
  float t = d[0] + d[1] + d[2] + d[3] + d[4] + d[5] + d[6] + d[7];
  return __shfl(t, 0, 32) + __shfl(t, 16, 32);
#else
  for (int off = 16; off > 0; off >>= 1) s += __shfl_down(s, off, 32);
  return __shfl(s, 0, 32);
#endif
}

// ---------------------------------------------------------------------------
// Pass 1: per-(channel, split) partial sums of {xr, xi, xr^2, xi^2, xr*xi}.
// Block (s*C + c) streams rows b = 2s, 2s+1 of channel c (ascending address
// order across the grid so L2 ends up holding the tail of the inputs).
// ---------------------------------------------------------------------------
__global__ void __launch_bounds__(THREADS)
cbn_stats(const float* __restrict__ xr, const float* __restrict__ xi,
          float* __restrict__ partial) {
  const int blk = blockIdx.x;
  const int sp  = blk / C_CH;
  const int ch  = blk % C_CH;
  const int t   = threadIdx.x;

  const v4f* xr4 = (const v4f*)xr;
  const v4f* xi4 = (const v4f*)xi;

  float s0 = 0.f, s1 = 0.f, s2 = 0.f, s3 = 0.f, s4 = 0.f;
#pragma unroll
  for (int r = 0; r < RPB; ++r) {
    const int  row  = sp * RPB + r;
    const long base = (long)(row * C_CH + ch) * F4ROW;
#pragma unroll
    for (int k = 0; k < 4; ++k) {
      const long idx = base + (long)(k * THREADS + t);
      v4f a = xr4[idx];
      v4f b = xi4[idx];
      s0 += a.x + a.y + a.z + a.w;
      s1 += b.x + b.y + b.z + b.w;
      s2 = fmaf(a.x, a.x, fmaf(a.y, a.y, fmaf(a.z, a.z, fmaf(a.w, a.w, s2))));
      s3 = fmaf(b.x, b.x, fmaf(b.y, b.y, fmaf(b.z, b.z, fmaf(b.w, b.w, s3))));
      s4 = fmaf(a.x, b.x, fmaf(a.y, b.y, fmaf(a.z, b.z, fmaf(a.w, b.w, s4))));
    }
  }

  // Wave reduction via WMMA (EXEC is all-ones: no divergence above).
  float w0 = wave_sum(s0);
  float w1 = wave_sum(s1);
  float w2 = wave_sum(s2);
  float w3 = wave_sum(s3);
  float w4 = wave_sum(s4);

  __shared__ float red[NWAVES][5];
  const int lane = t & 31;
  const int wv   = t >> 5;
  if (lane == 0) {
    red[wv][0] = w0; red[wv][1] = w1; red[wv][2] = w2;
    red[wv][3] = w3; red[wv][4] = w4;
  }
  __syncthreads();
  if (t < 5) {
    float acc = 0.f;
#pragma unroll
    for (int w = 0; w < NWAVES; ++w) acc += red[w][t];
    partial[(long)blk * 5 + t] = acc;
  }
}

// ---------------------------------------------------------------------------
// Pass 2: fold partials -> per-channel fused coefficients (1 block, 256 thr).
//   out_r = Crr*xr + Cri*xi + offr ; out_i = Dir*xr + Dii*xi + offi
// ---------------------------------------------------------------------------
__global__ void __launch_bounds__(C_CH)
cbn_coeff(const float* __restrict__ partial, const float* __restrict__ weight,
          const float* __restrict__ bias, float* __restrict__ coeff) {
  const int c = threadIdx.x;
  float S0 = 0.f, S1 = 0.f, S2 = 0.f, S3 = 0.f, S4 = 0.f;
  for (int s = 0; s < SPLIT; ++s) {
    const float* p = partial + (long)(s * C_CH + c) * 5;
    S0 += p[0]; S1 += p[1]; S2 += p[2]; S3 += p[3]; S4 += p[4];
  }
  const float invn = 1.0f / (float)(BATCH * HW);
  const float mur = S0 * invn, mui = S1 * invn;
  const float Vrr = fmaf(-mur, mur, S2 * invn) + EPSV;
  const float Vii = fmaf(-mui, mui, S3 * invn) + EPSV;
  const float Vri = fmaf(-mur, mui, S4 * invn);

  const float sd  = sqrtf(Vrr * Vii - Vri * Vri);
  const float tr  = sqrtf(Vrr + Vii + 2.0f * sd);
  const float ist = 1.0f / (sd * tr);
  const float Rrr = (Vii + sd) * ist;
  const float Rii = (Vrr + sd) * ist;
  const float Rri = -Vri * ist;

  const float w00 = weight[0 * C_CH + c], w01 = weight[1 * C_CH + c];
  const float w10 = weight[2 * C_CH + c], w11 = weight[3 * C_CH + c];
  const float b0  = bias[0 * C_CH + c],   b1  = bias[1 * C_CH + c];

  const float Crr = fmaf(w00, Rrr, w01 * Rri);
  const float Cri = fmaf(w00, Rri, w01 * Rii);
  const float Dir = fmaf(w10, Rrr, w11 * Rri);
  const float Dii = fmaf(w10, Rri, w11 * Rii);
  const float offr = b0 - Crr * mur - Cri * mui;
  const float offi = b1 - Dir * mur - Dii * mui;

  float* o = coeff + (long)c * 6;
  o[0] = Crr; o[1] = Cri; o[2] = offr;
  o[3] = Dir; o[4] = Dii; o[5] = offi;
}

// ---------------------------------------------------------------------------
// Pass 3: elementwise apply.  Blocks walk memory in REVERSE order so the input
// tail that pass 1 left resident in the 192 MB L2 is re-used before eviction.
// Output is write-once -> non-temporal b128 stores keep it out of L2.
// ---------------------------------------------------------------------------
__global__ void __launch_bounds__(THREADS)
cbn_apply(const float* __restrict__ xr, const float* __restrict__ xi,
          const float* __restrict__ coeff, float* __restrict__ out) {
  const int nblk = BATCH * C_CH;
  const int rblk = nblk - 1 - (int)blockIdx.x;  // reverse for L2 reuse
  const int ch   = rblk % C_CH;
  const int t    = threadIdx.x;

  const float* co = coeff + (long)ch * 6;
  const float Crr = co[0], Cri = co[1], offr = co[2];
  const float Dir = co[3], Dii = co[4], offi = co[5];

  const long base = (long)rblk * F4ROW;
  const v4f* xr4 = (const v4f*)xr;
  const v4f* xi4 = (const v4f*)xi;
  v4f* or4 = (v4f*)out;
  v4f* oi4 = (v4f*)(out + (long)BATCH * C_CH * HW);

#pragma unroll
  for (int k = 0; k < 4; ++k) {
    const long idx = base + (long)(k * THREADS + t);
    v4f a = xr4[idx];
    v4f b = xi4[idx];
    v4f r, m;
    r.x = fmaf(Crr, a.x, fmaf(Cri, b.x, offr));
    r.y = fmaf(Crr, a.y, fmaf(Cri, b.y, offr));
    r.z = fmaf(Crr, a.z, fmaf(Cri, b.z, offr));
    r.w = fmaf(Crr, a.w, fmaf(Cri, b.w, offr));
    m.x = fmaf(Dir, a.x, fmaf(Dii, b.x, offi));
    m.y = fmaf(Dir, a.y, fmaf(Dii, b.y, offi));
    m.z = fmaf(Dir, a.z, fmaf(Dii, b.z, offi));
    m.w = fmaf(Dir, a.w, fmaf(Dii, b.w, offi));
    __builtin_nontemporal_store(r, &or4[idx]);
    __builtin_nontemporal_store(m, &oi4[idx]);
  }
}

// ---------------------------------------------------------------------------
extern "C" void kernel_launch(void* const* d_in, const int* in_sizes, int n_in,
                              void* d_out, int out_size, void* d_ws,
                              size_t ws_size, hipStream_t stream) {
  const float* xr     = (const float*)d_in[0];  // [B,C,H,W]
  const float* xi     = (const float*)d_in[1];  // [B,C,H,W]
  const float* weight = (const float*)d_in[2];  // [2,2,C]
  const float* bias   = (const float*)d_in[3];  // [2,C]
  float*       out    = (float*)d_out;          // [2,B,C,H,W]

  float* partial = (float*)d_ws;                       // SPLIT*C*5 floats
  float* coeff   = partial + (long)SPLIT * C_CH * 5;   // C*6 floats

  cbn_stats<<<SPLIT * C_CH, THREADS, 0, stream>>>(xr, xi, partial);
  cbn_coeff<<<1, C_CH, 0, stream>>>(partial, weight, bias, coeff);
  cbn_apply<<<BATCH * C_CH, THREADS, 0, stream>>>(xr, xi, coeff, out);
}